// CausalMHA_78915729096792
// MI455X (gfx1250) — compile-verified
//
#include <hip/hip_runtime.h>
#include <math.h>

#define BATCH 2
#define SEQ   2048
#define DM    1024
#define NH    16
#define DH    64
#define WIN   512
#define NW    4   // waves per block

typedef float v2f __attribute__((ext_vector_type(2)));
typedef float v8f __attribute__((ext_vector_type(8)));

__device__ __forceinline__ v8f wmma_f32(v2f a, v2f b, v8f c) {
  // D = A(16x4) * B(4x16) + C(16x16), all f32
  return __builtin_amdgcn_wmma_f32_16x16x4_f32(false, a, false, b, (short)0, c,
                                               false, false);
}

// ---------------------------------------------------------------------------
// C[M,N] = A[M,K] @ W[N,K]^T   (row-major A, row-major W as (out,in))
// Wave computes a 32x64 tile: 2 A-fragments x 4 B-fragments -> 8 WMMAs per
// K-step of 4. Fragments use a 2-deep register pipeline: loads for K-step
// k+8 are issued while K-step k computes, so loadcnt waits have ~16 WMMAs
// of slack (covers L2 latency on the cold pass).
// ---------------------------------------------------------------------------
__global__ __launch_bounds__(128) void gemm_nt_f32_wmma(
    const float* __restrict__ A, const float* __restrict__ W,
    float* __restrict__ C, int M, int N, int K) {
  const int lane = threadIdx.x & 31;
  const int wave = threadIdx.x >> 5;
  const int half = lane >> 4;      // 0: lanes 0-15 (K=0,1)  1: lanes 16-31 (K=2,3)
  const int lidx = lane & 15;
  const int m0 = blockIdx.x * 32;
  const int n0 = (blockIdx.y * NW + wave) * 64;

  const float* Ap0 = A + (size_t)(m0 + lidx) * K + 2 * half;
  const float* Ap1 = Ap0 + (size_t)16 * K;
  const float* Bp  = W + (size_t)(n0 + lidx) * K + 2 * half;

  // warm L2/L1 ahead of the stream (global_prefetch_b8)
  __builtin_prefetch(Ap0 + 128, 0, 1);
  __builtin_prefetch(Ap1 + 128, 0, 1);
  __builtin_prefetch(Bp + 128, 0, 1);
  __builtin_prefetch(Bp + (size_t)16 * K + 128, 0, 1);
  __builtin_prefetch(Bp + (size_t)32 * K + 128, 0, 1);
  __builtin_prefetch(Bp + (size_t)48 * K + 128, 0, 1);

  v8f acc[8];
#pragma unroll
  for (int i = 0; i < 8; ++i) acc[i] = (v8f){0, 0, 0, 0, 0, 0, 0, 0};

  // 2-deep pipeline prologue: stage 0 (k=0) and stage 1 (k=4)
  v2f a0 = *(const v2f*)(Ap0);
  v2f a1 = *(const v2f*)(Ap1);
  v2f b0 = *(const v2f*)(Bp);
  v2f b1 = *(const v2f*)(Bp + (size_t)16 * K);
  v2f b2 = *(const v2f*)(Bp + (size_t)32 * K);
  v2f b3 = *(const v2f*)(Bp + (size_t)48 * K);
  v2f pa0 = *(const v2f*)(Ap0 + 4);
  v2f pa1 = *(const v2f*)(Ap1 + 4);
  v2f pb0 = *(const v2f*)(Bp + 4);
  v2f pb1 = *(const v2f*)(Bp + (size_t)16 * K + 4);
  v2f pb2 = *(const v2f*)(Bp + (size_t)32 * K + 4);
  v2f pb3 = *(const v2f*)(Bp + (size_t)48 * K + 4);

#pragma unroll 4
  for (int k = 8; k < K; k += 4) {
    // issue loads two K-steps ahead of the WMMAs that consume them
    v2f na0 = *(const v2f*)(Ap0 + k);
    v2f na1 = *(const v2f*)(Ap1 + k);
    v2f nb0 = *(const v2f*)(Bp + k);
    v2f nb1 = *(const v2f*)(Bp + (size_t)16 * K + k);
    v2f nb2 = *(const v2f*)(Bp + (size_t)32 * K + k);
    v2f nb3 = *(const v2f*)(Bp + (size_t)48 * K + k);
    acc[0] = wmma_f32(a0, b0, acc[0]);
    acc[1] = wmma_f32(a0, b1, acc[1]);
    acc[2] = wmma_f32(a0, b2, acc[2]);
    acc[3] = wmma_f32(a0, b3, acc[3]);
    acc[4] = wmma_f32(a1, b0, acc[4]);
    acc[5] = wmma_f32(a1, b1, acc[5]);
    acc[6] = wmma_f32(a1, b2, acc[6]);
    acc[7] = wmma_f32(a1, b3, acc[7]);
    a0 = pa0; a1 = pa1; b0 = pb0; b1 = pb1; b2 = pb2; b3 = pb3;
    pa0 = na0; pa1 = na1; pb0 = nb0; pb1 = nb1; pb2 = nb2; pb3 = nb3;
  }
  // epilogue: two staged K-steps remain
  acc[0] = wmma_f32(a0, b0, acc[0]);
  acc[1] = wmma_f32(a0, b1, acc[1]);
  acc[2] = wmma_f32(a0, b2, acc[2]);
  acc[3] = wmma_f32(a0, b3, acc[3]);
  acc[4] = wmma_f32(a1, b0, acc[4]);
  acc[5] = wmma_f32(a1, b1, acc[5]);
  acc[6] = wmma_f32(a1, b2, acc[6]);
  acc[7] = wmma_f32(a1, b3, acc[7]);
  acc[0] = wmma_f32(pa0, pb0, acc[0]);
  acc[1] = wmma_f32(pa0, pb1, acc[1]);
  acc[2] = wmma_f32(pa0, pb2, acc[2]);
  acc[3] = wmma_f32(pa0, pb3, acc[3]);
  acc[4] = wmma_f32(pa1, pb0, acc[4]);
  acc[5] = wmma_f32(pa1, pb1, acc[5]);
  acc[6] = wmma_f32(pa1, pb2, acc[6]);
  acc[7] = wmma_f32(pa1, pb3, acc[7]);

  // C/D layout: VGPR r -> row +r + 8*half, col n0 + lidx (+16 per B-frag)
#pragma unroll
  for (int i = 0; i < 2; ++i) {
    float* Cp = C + (size_t)(m0 + 16 * i + 8 * half) * N + n0 + lidx;
#pragma unroll
    for (int r = 0; r < 8; ++r) {
      Cp[(size_t)r * N +  0] = acc[4 * i + 0][r];
      Cp[(size_t)r * N + 16] = acc[4 * i + 1][r];
      Cp[(size_t)r * N + 32] = acc[4 * i + 2][r];
      Cp[(size_t)r * N + 48] = acc[4 * i + 3][r];
    }
  }
}

// ---------------------------------------------------------------------------
// In-place rotary on q and k portions of the qkv buffer (B, L, 3*DM).
// One thread handles one (b,l,h,i) with i in [0,16): pair (i, i+16).
// Precise sincosf/powf: angles reach ~2047 rad, fast-native range reduction
// would be inaccurate there.
// ---------------------------------------------------------------------------
__global__ void rotary_qk_inplace(float* __restrict__ qkv) {
  int idx = blockIdx.x * blockDim.x + threadIdx.x;
  int i  = idx & 15;
  int h  = (idx >> 4) & (NH - 1);
  int bl = idx >> 8;
  if (bl >= BATCH * SEQ) return;
  float pos = (float)(bl & (SEQ - 1));
  float inv_freq = powf(10000.0f, -((float)i) * (1.0f / 16.0f));
  float ang = pos * inv_freq;
  float s, c;
  sincosf(ang, &s, &c);

  size_t base = (size_t)bl * (3 * DM) + h * DH;
  float x1 = qkv[base + i], x2 = qkv[base + i + 16];
  qkv[base + i]      = x1 * c - x2 * s;
  qkv[base + i + 16] = x2 * c + x1 * s;

  size_t kb = base + DM;
  x1 = qkv[kb + i]; x2 = qkv[kb + i + 16];
  qkv[kb + i]      = x1 * c - x2 * s;
  qkv[kb + i + 16] = x2 * c + x1 * s;
}

// ---------------------------------------------------------------------------
// Flash attention, f32 WMMA. One wave per (b, h, 16-query tile).
// qkv layout (B, L, 3*DM): q at +0, k at +DM, v at +2*DM within a row.
// ctx output layout (B, L, DM) with head-major columns (h*DH + d).
// All fragments for a tile are loaded before the WMMA chains so VMEM
// latency overlaps with the matrix pipe.
// ---------------------------------------------------------------------------
__global__ __launch_bounds__(128) void flash_attn_f32_wmma(
    const float* __restrict__ qkv, const unsigned char* __restrict__ msk,
    float* __restrict__ ctx) {
  __shared__ float plds[NW][16][18];  // padded 16x16 P tile per wave

  const int lane = threadIdx.x & 31;
  const int wave = threadIdx.x >> 5;
  const int half = lane >> 4;
  const int lidx = lane & 15;
  const int tile = blockIdx.x * NW + wave;
  const int qt = tile & (SEQ / 16 - 1);
  const int h  = (tile >> 7) & (NH - 1);
  const int b  = tile >> 11;
  const int q0 = qt * 16;
  const size_t row_stride = 3 * DM;

  // Preload Q A-fragments: 16 chunks of K=4 covering DH=64
  const float* qp =
      qkv + (size_t)(b * SEQ + q0 + lidx) * row_stride + h * DH + 2 * half;
  v2f qa[16];
#pragma unroll
  for (int t = 0; t < 16; ++t) qa[t] = *(const v2f*)(qp + 4 * t);

  v8f o[4];
  float mrow[8], lrow[8];
#pragma unroll
  for (int t = 0; t < 4; ++t) o[t] = (v8f){0, 0, 0, 0, 0, 0, 0, 0};
#pragma unroll
  for (int r = 0; r < 8; ++r) { mrow[r] = -1e30f; lrow[r] = 0.0f; }

  int kstart = q0 - (WIN - 1);
  if (kstart < 0) kstart = 0;
  const int kt0 = kstart >> 4;

  for (int kt = kt0; kt <= qt; ++kt) {
    const int kbase = kt * 16;
    const float* kp = qkv + (size_t)(b * SEQ + kbase + lidx) * row_stride +
                      DM + h * DH + 2 * half;
    const float* vp = qkv + (size_t)(b * SEQ + kbase) * row_stride +
                      2 * DM + h * DH + lidx;

    // prefetch next key tile's K/V rows while this tile computes
    if (kt < qt) {
      __builtin_prefetch(kp + 16 * row_stride, 0, 1);
      __builtin_prefetch(vp + 16 * row_stride, 0, 1);
    }

    // --- load all 16 K-fragments, then 16 WMMAs: S = Q K^T ---
    v2f kb[16];
#pragma unroll
    for (int t = 0; t < 16; ++t) kb[t] = *(const v2f*)(kp + 4 * t);
    v8f s = {0, 0, 0, 0, 0, 0, 0, 0};
#pragma unroll
    for (int t = 0; t < 16; ++t) s = wmma_f32(qa[t], kb[t], s);

    // --- mask + online softmax (rows live across 16-lane halves) ---
    const bool keyok = msk[(size_t)b * SEQ + kbase + lidx] != 0;
    const int kcol = kbase + lidx;
#pragma unroll
    for (int r = 0; r < 8; ++r) {
      const int qrow = q0 + r + 8 * half;
      float sv = s[r] * 0.125f;  // HEAD_DIM^-0.5
      bool ok = keyok && (kcol <= qrow) && (kcol >= qrow - (WIN - 1));
      sv = ok ? sv : -1e30f;

      float rmax = sv;
#pragma unroll
      for (int off = 8; off; off >>= 1) rmax = fmaxf(rmax, __shfl_xor(rmax, off));
      float mnew  = fmaxf(mrow[r], rmax);
      float alpha = expf(mrow[r] - mnew);
      float pr    = expf(sv - mnew);
      float rsum = pr;
#pragma unroll
      for (int off = 8; off; off >>= 1) rsum += __shfl_xor(rsum, off);
      lrow[r] = lrow[r] * alpha + rsum;
      mrow[r] = mnew;
#pragma unroll
      for (int t = 0; t < 4; ++t) o[t][r] *= alpha;
      plds[wave][r + 8 * half][lidx] = pr;  // stage P for A-fragment reload
    }

    // --- O += P V : load all P and V fragments, then 16 WMMAs ---
    v2f pa[4];
#pragma unroll
    for (int c = 0; c < 4; ++c)
      pa[c] = *(const v2f*)&plds[wave][lidx][4 * c + 2 * half];
    v2f vb[4][4];
#pragma unroll
    for (int c = 0; c < 4; ++c) {
      const int krow = 4 * c + 2 * half;
#pragma unroll
      for (int t = 0; t < 4; ++t) {
        vb[c][t].x = vp[(size_t)(krow)     * row_stride + t * 16];
        vb[c][t].y = vp[(size_t)(krow + 1) * row_stride + t * 16];
      }
    }
#pragma unroll
    for (int c = 0; c < 4; ++c)
#pragma unroll
      for (int t = 0; t < 4; ++t) o[t] = wmma_f32(pa[c], vb[c][t], o[t]);
  }

  // normalize and store ctx
#pragma unroll
  for (int r = 0; r < 8; ++r) {
    const float invl = 1.0f / fmaxf(lrow[r], 1e-20f);
    float* cp = ctx + (size_t)(b * SEQ + q0 + r + 8 * half) * DM + h * DH + lidx;
#pragma unroll
    for (int t = 0; t < 4; ++t) cp[t * 16] = o[t][r] * invl;
  }
}

// ---------------------------------------------------------------------------
extern "C" void kernel_launch(void* const* d_in, const int* in_sizes, int n_in,
                              void* d_out, int out_size, void* d_ws,
                              size_t ws_size, hipStream_t stream) {
  (void)in_sizes; (void)n_in; (void)out_size; (void)ws_size;
  const float* x             = (const float*)d_in[0];
  const unsigned char* amask = (const unsigned char*)d_in[1];
  const float* Wqkv          = (const float*)d_in[2];
  const float* Wout          = (const float*)d_in[3];
  float* out = (float*)d_out;

  float* qkv = (float*)d_ws;                              // B*L*3*DM floats (48 MB)
  float* ctx = qkv + (size_t)BATCH * SEQ * 3 * DM;        // B*L*DM floats  (16 MB)

  const int M = BATCH * SEQ;  // 4096
  dim3 blk(32 * NW);

  // 1) qkv = x @ Wqkv^T          (4096 x 3072, K=1024)
  gemm_nt_f32_wmma<<<dim3(M / 32, (3 * DM) / (64 * NW)), blk, 0, stream>>>(
      x, Wqkv, qkv, M, 3 * DM, DM);

  // 2) rotary on q,k in place
  rotary_qk_inplace<<<(BATCH * SEQ * NH * 16) / 256, 256, 0, stream>>>(qkv);

  // 3) windowed causal flash attention -> ctx
  flash_attn_f32_wmma<<<(BATCH * NH * (SEQ / 16)) / NW, blk, 0, stream>>>(
      qkv, amask, ctx);

  // 4) out = ctx @ Wout^T        (4096 x 1024, K=1024)
  gemm_nt_f32_wmma<<<dim3(M / 32, DM / (64 * NW)), blk, 0, stream>>>(
      ctx, Wout, out, M, DM, DM);
}